// NP_Loss_12945031430627
// MI455X (gfx1250) — compile-verified
//
#include <hip/hip_runtime.h>
#include <hip/hip_bf16.h>
#include <math.h>

// ---------------- problem constants ----------------
#define BATCH   16384
#define KDIM    576          // 24*24
#define HDIM    24
#define LAMDA   0.01f

// ---------------- gram kernel tiling ----------------
#define STRIDE_A 578         // A tile stride: bank stride 2, b64-aligned
#define STRIDE_B 580         // B tile stride: bank stride 4, b64-aligned
#define BLOCK_I  96          // rows of E per workgroup (6 waves x 16)
#define BLOCK_J  32          // J columns per iteration (2 subtiles of 16)
#define NWAVES   6
#define NTHREADS (NWAVES * 32)
#define GRAM_LDS_FLOATS (BLOCK_I * STRIDE_A + BLOCK_J * STRIDE_B)
#define GRAM_LDS_BYTES  (GRAM_LDS_FLOATS * 4)     // 296,192 B <= 320 KB

typedef float v2f __attribute__((ext_vector_type(2)));
typedef float v8f __attribute__((ext_vector_type(8)));

// ---------------------------------------------------------------------------
// 0) zero the accumulators: ws[0] = pair-loss acc, ws[1..576] = psd acc
// ---------------------------------------------------------------------------
__global__ void zero_kernel(float* __restrict__ ws) {
    int t = threadIdx.x;
    if (t < 577) ws[t] = 0.0f;
}

// ---------------------------------------------------------------------------
// 1) per-row squared norms: one wave32 per row, coalesced, shfl reduction
// ---------------------------------------------------------------------------
__global__ __launch_bounds__(256) void sqnorm_kernel(const float* __restrict__ E,
                                                     float* __restrict__ sq) {
    int row  = blockIdx.x * (blockDim.x >> 5) + (threadIdx.x >> 5);
    int lane = threadIdx.x & 31;
    if (row >= BATCH) return;
    const float* p = E + (size_t)row * KDIM;
    float acc = 0.0f;
    for (int k = lane; k < KDIM; k += 32) {
        float v = p[k];
        acc = fmaf(v, v, acc);
    }
    for (int off = 1; off < 32; off <<= 1) acc += __shfl_xor(acc, off, 32);
    if (lane == 0) sq[row] = acc;
}

// ---------------------------------------------------------------------------
// 2) PSD accumulation: 576 threads per block (one output freq (u,v) each),
//    32 images per block, twiddle tables in LDS, atomicAdd partial |X|^2.
// ---------------------------------------------------------------------------
#define IMGS_PER_BLOCK 32
__global__ __launch_bounds__(576) void psd_kernel(const float* __restrict__ E,
                                                  float* __restrict__ psd) {
    __shared__ float img[KDIM];
    __shared__ float ct[HDIM], st[HDIM];
    int t = threadIdx.x;
    if (t < HDIM) {
        float a = 6.28318530718f * (float)t / (float)HDIM;
        ct[t] = __cosf(a);
        st[t] = __sinf(a);   // sign of Im irrelevant for |X|^2
    }
    int u = t / HDIM, v = t % HDIM;
    int vxm[HDIM];
#pragma unroll
    for (int x = 0; x < HDIM; ++x) vxm[x] = (v * x) % HDIM;

    float acc = 0.0f;
    int base = blockIdx.x * IMGS_PER_BLOCK;
    for (int im = 0; im < IMGS_PER_BLOCK; ++im) {
        __syncthreads();
        img[t] = E[(size_t)(base + im) * KDIM + t];
        __syncthreads();
        float re = 0.0f, imp = 0.0f;
        for (int y = 0; y < HDIM; ++y) {
            int uy = (u * y) % HDIM;
            const float* row = img + y * HDIM;
#pragma unroll
            for (int x = 0; x < HDIM; ++x) {
                int m = uy + vxm[x];
                if (m >= HDIM) m -= HDIM;
                float p = row[x];
                re  = fmaf(p, ct[m], re);
                imp = fmaf(p, st[m], imp);
            }
        }
        acc += re * re + imp * imp;
    }
    atomicAdd(psd + t, acc);
}

// ---------------------------------------------------------------------------
// 3) fused Gram / distance / streaming-softmax kernel (the 309-GFLOP part).
//    6 waves, 96 I-rows resident in LDS; J streams in 32-col tiles.
//    Each wave: two 16x16 f32 subtiles sharing one A fragment per K-step
//    (DS ops per WMMA: 1.5 instead of 2.0). Streaming softmax keeps only
//    S = sum exp(-d) and label-masked T per row (exponents <= 0: no shift).
// ---------------------------------------------------------------------------
__global__ __launch_bounds__(NTHREADS) void gram_kernel(const float* __restrict__ E,
                                                        const int*   __restrict__ labels,
                                                        const float* __restrict__ sq,
                                                        float*       __restrict__ pair_acc) {
    extern __shared__ float smem[];
    float* sA = smem;                          // [BLOCK_I][STRIDE_A]
    float* sB = smem + BLOCK_I * STRIDE_A;     // [BLOCK_J][STRIDE_B]

    const int tid  = threadIdx.x;
    const int lane = tid & 31;
    const int wave = tid >> 5;
    const int half = lane >> 4;                // 0: low half, 1: high half
    const int lcol = lane & 15;                // N within subtile
    const int i0   = blockIdx.x * BLOCK_I;

    // ---- stage A strip (BLOCK_I x KDIM) into LDS with padded stride ----
    {
        const int nf2 = KDIM / 2;              // 288 float2 per row
        for (int idx = tid; idx < BLOCK_I * nf2; idx += NTHREADS) {
            int r = idx / nf2, c2 = idx - r * nf2;
            int gr = i0 + r;
            if (gr >= BATCH) gr = 0;           // tail pad; outputs guarded below
            float2 val = *(const float2*)(E + (size_t)gr * KDIM + c2 * 2);
            *(float2*)(sA + r * STRIDE_A + c2 * 2) = val;
        }
    }

    // ---- per-lane row metadata for the C/D striping (M = v + 8*half) ----
    float sqi[8];
    int   labi[8], growi[8];
#pragma unroll
    for (int v = 0; v < 8; ++v) {
        int gr = i0 + wave * 16 + v + 8 * half;
        growi[v] = gr;
        int grc  = gr < BATCH ? gr : 0;
        sqi[v]  = sq[grc];
        labi[v] = labels[grc];
    }

    float Ssum[8], Tsum[8];
#pragma unroll
    for (int v = 0; v < 8; ++v) { Ssum[v] = 0.0f; Tsum[v] = 0.0f; }

    // A frag: lane holds row (lane&15) of the wave strip, K pair k0+2*half
    const float* aBase  = sA + (wave * 16 + lcol) * STRIDE_A + 2 * half;
    // B frags: subtile 0 = sB rows 0..15, subtile 1 = sB rows 16..31
    const float* b0Base = sB + lcol * STRIDE_B + 2 * half;
    const float* b1Base = sB + (16 + lcol) * STRIDE_B + 2 * half;

    for (int j0 = 0; j0 < BATCH; j0 += BLOCK_J) {
        __syncthreads();
        // stage B tile: 32 rows of E (the J columns), padded stride
        {
            const int nf2 = KDIM / 2;
            for (int idx = tid; idx < BLOCK_J * nf2; idx += NTHREADS) {
                int r = idx / nf2, c2 = idx - r * nf2;
                float2 val = *(const float2*)(E + (size_t)(j0 + r) * KDIM + c2 * 2);
                *(float2*)(sB + r * STRIDE_B + c2 * 2) = val;
            }
        }
        __syncthreads();

        v8f C0 = {}, C1 = {};
#pragma unroll 4
        for (int k0 = 0; k0 < KDIM; k0 += 4) {
            v2f a  = *(const v2f*)(aBase  + k0);
            v2f b0 = *(const v2f*)(b0Base + k0);
            v2f b1 = *(const v2f*)(b1Base + k0);
            C0 = __builtin_amdgcn_wmma_f32_16x16x4_f32(
                     false, a, false, b0, (short)0, C0, false, false);
            C1 = __builtin_amdgcn_wmma_f32_16x16x4_f32(
                     false, a, false, b1, (short)0, C1, false, false);
        }

        // ---- fold both subtiles into streaming softmax accumulators ----
        // (branchless: selects compile to v_cndmask, no exec-mask churn)
        float sqj0  = sq[j0 + lcol];
        float sqj1  = sq[j0 + 16 + lcol];
        int   labj0 = labels[j0 + lcol];
        int   labj1 = labels[j0 + 16 + lcol];
#pragma unroll
        for (int v = 0; v < 8; ++v) {
            float d20 = fmaxf(sqi[v] + sqj0 - 2.0f * C0[v], 0.0f);
            float w0  = __expf(-__fsqrt_rn(d20));
            w0 = (growi[v] == j0 + lcol) ? 0.0f : w0;          // drop diagonal
            Ssum[v] += w0;
            Tsum[v] += (labi[v] == labj0) ? w0 : 0.0f;

            float d21 = fmaxf(sqi[v] + sqj1 - 2.0f * C1[v], 0.0f);
            float w1  = __expf(-__fsqrt_rn(d21));
            w1 = (growi[v] == j0 + 16 + lcol) ? 0.0f : w1;
            Ssum[v] += w1;
            Tsum[v] += (labi[v] == labj1) ? w1 : 0.0f;
        }
    }

    // ---- reduce each row's 16 column-lanes (stay within the 16-lane half) ----
#pragma unroll
    for (int v = 0; v < 8; ++v) {
        float s = Ssum[v], t = Tsum[v];
        for (int off = 1; off < 16; off <<= 1) {
            s += __shfl_xor(s, off, 32);
            t += __shfl_xor(t, off, 32);
        }
        if (lcol == 0 && growi[v] < BATCH) {
            // -log(prob) = log S - log T
            atomicAdd(pair_acc, __logf(s) - __logf(t));
        }
    }
}

// ---------------------------------------------------------------------------
// 4) finalize: spectral-flatness clip + pair term -> scalar loss
// ---------------------------------------------------------------------------
__global__ __launch_bounds__(576) void finalize_kernel(const float* __restrict__ psd,
                                                       const float* __restrict__ pair_acc,
                                                       float* __restrict__ out) {
    __shared__ float rlog[KDIM];
    __shared__ float rsum[KDIM];
    int t = threadIdx.x;
    float a = psd[t] / (float)BATCH;
    rlog[t] = __logf(a);
    rsum[t] = a;
    __syncthreads();
    if (t == 0) {
        float sl = 0.0f, s = 0.0f;
        for (int i = 0; i < KDIM; ++i) { sl += rlog[i]; s += rsum[i]; }
        float lp = sl / (float)KDIM - __logf(s / (float)KDIM);
        lp = fminf(fmaxf(lp, 0.0f), 100.0f);
        out[0] = -LAMDA * lp + pair_acc[0] / 200.0f;
    }
}

// ---------------------------------------------------------------------------
extern "C" void kernel_launch(void* const* d_in, const int* in_sizes, int n_in,
                              void* d_out, int out_size, void* d_ws, size_t ws_size,
                              hipStream_t stream) {
    const float* E      = (const float*)d_in[0];   // (16384,1,24,24) fp32
    const int*   labels = (const int*)d_in[1];     // (16384,) int
    float*       out    = (float*)d_out;           // scalar loss
    float*       ws     = (float*)d_ws;

    float* pair = ws;          // ws[0]
    float* psd  = ws + 1;      // ws[1..576]
    float* sq   = ws + 1024;   // ws[1024 .. 1024+16384)

    // 0) zero accumulators
    zero_kernel<<<1, 640, 0, stream>>>(ws);

    // 1) row squared norms (one wave per row, 8 waves/block)
    sqnorm_kernel<<<(BATCH + 7) / 8, 256, 0, stream>>>(E, sq);

    // 2) PSD accumulation
    psd_kernel<<<BATCH / IMGS_PER_BLOCK, 576, 0, stream>>>(E, psd);

    // 3) fused Gram + streaming softmax (fp32 WMMA)
    int nblk = (BATCH + BLOCK_I - 1) / BLOCK_I;    // 171
    gram_kernel<<<nblk, NTHREADS, GRAM_LDS_BYTES, stream>>>(E, labels, sq, pair);

    // 4) finalize scalar
    finalize_kernel<<<1, KDIM, 0, stream>>>(psd, pair, out);
}